// LocBlock1dNT_75196287419027
// MI455X (gfx1250) — compile-verified
//
#include <hip/hip_runtime.h>

// LocBlock1dNT: per-patch GEMM  y[n,o,p] = relu( (1/8) * sum_{c,f} x[n,c,p*4+f] * w[o,c,p,f] )
// M=8192 (n), N=32 (o), K=64 (c*4+f), 512 independent patches.
// fp32 end-to-end via V_WMMA_F32_16X16X4_F32 (16 chained WMMAs per 16x16 tile).
// Memory-bound: ~1.61 GB traffic -> ~69 us floor at 23.3 TB/s; WMMA keeps math far under that.

typedef __attribute__((ext_vector_type(2))) float v2f;
typedef __attribute__((ext_vector_type(4))) float v4f;
typedef __attribute__((ext_vector_type(8))) float v8f;

namespace {
constexpr int kN  = 8192;
constexpr int kCI = 16;
constexpr int kCO = 32;
constexpr int kP  = 512;
constexpr int kFS = 4;
constexpr int kRowStride = kCI * kP * kFS; // 32768 floats: stride of one n-row of x AND one o-row of w
constexpr int kPB = 32;                    // patches per workgroup (=> 128B-coalesced output rows)
constexpr int kPW = 8;                     // patches per wave (4 waves / workgroup)
constexpr int kSP = 36;                    // padded p-stride in LDS (floats): 144B = 16B aligned, bank-friendly
}

__global__ __launch_bounds__(128)
void LocBlock1dNT_wmma_kernel(const float* __restrict__ x,
                              const float* __restrict__ w,
                              float* __restrict__ y) {
  __shared__ float lds[16 * kCO * kSP];     // 72 KB staging for 16 x 32 x 32 output block

  const int tid    = threadIdx.x;
  const int wave   = tid >> 5;
  const int lane   = tid & 31;
  const int laneLo = lane & 15;
  const int laneHi = lane >> 4;             // 0: K={0,1}/M rows 0-7 side, 1: K={2,3}/M rows 8-15 side

  const int n0 = blockIdx.y * 16;           // 16-row batch block
  const int p0 = blockIdx.x * kPB;          // 32-patch block

  // A fragment source: lane holds x[n0+laneLo, c, p*4 + 2*laneHi + {0,1}]  (b64 per chunk)
  const float* xb  = x + (size_t)(n0 + laneLo) * kRowStride + laneHi * 2;
  // B fragment source: lane holds w[otile*16+laneLo, c, p, 2*laneHi + {0,1}]
  const float* wb0 = w + (size_t)laneLo * kRowStride + laneHi * 2;
  const float* wb1 = wb0 + (size_t)16 * kRowStride;

  #pragma unroll 1
  for (int pi = 0; pi < kPW; ++pi) {
    const int pofs = wave * kPW + pi;       // patch slot within workgroup block
    const int p    = p0 + pofs;
    const int pel  = p * kFS;

    v2f a[kCI], b0[kCI], b1[kCI];
    #pragma unroll
    for (int c = 0; c < kCI; ++c) {
      a[c]  = *(const v2f*)(xb  + c * (kP * kFS) + pel);
      b0[c] = *(const v2f*)(wb0 + c * (kP * kFS) + pel);
      b1[c] = *(const v2f*)(wb1 + c * (kP * kFS) + pel);
    }

    v8f acc0 = {};
    v8f acc1 = {};
    #pragma unroll
    for (int c = 0; c < kCI; ++c) {
      // K=64 as 16 chained 16x16x4 f32 WMMAs, two N-tiles (o 0-15, o 16-31)
      acc0 = __builtin_amdgcn_wmma_f32_16x16x4_f32(false, a[c], false, b0[c],
                                                   (short)0, acc0, false, false);
      acc1 = __builtin_amdgcn_wmma_f32_16x16x4_f32(false, a[c], false, b1[c],
                                                   (short)0, acc1, false, false);
    }

    // epilogue: relu(y/8) = max(0.125*acc, 0); stage to LDS at [m][o][pofs]
    #pragma unroll
    for (int g = 0; g < 8; ++g) {
      const int m = g + 8 * laneHi;         // C/D layout: VGPR g, lanes 0-15 -> M=g, lanes 16-31 -> M=8+g
      const float t0 = fmaxf(acc0[g] * 0.125f, 0.0f);
      const float t1 = fmaxf(acc1[g] * 0.125f, 0.0f);
      lds[(m * kCO + laneLo)        * kSP + pofs] = t0;
      lds[(m * kCO + 16 + laneLo)   * kSP + pofs] = t1;
    }
  }

  __syncthreads();

  // Cooperative coalesced write-out: p fastest, float4 per lane.
  // 8 consecutive lanes cover one (n,o) row of 32 floats = one full 128B cacheline.
  for (int idx = tid; idx < 16 * kCO * (kPB / 4); idx += 128) {
    const int p4  = idx & (kPB / 4 - 1);    // float4 slot within row (0..7)
    const int row = idx >> 3;               // n*32 + o
    const int o   = row & (kCO - 1);
    const int nn  = row >> 5;
    v4f v = *(const v4f*)&lds[row * kSP + p4 * 4];
    float* dst = y + (((size_t)(n0 + nn) * kCO + o) * kP + p0 + p4 * 4);
    __builtin_nontemporal_store(v, (v4f*)dst);  // output (536MB) has no reuse; bypass cache rinse
  }
}

extern "C" void kernel_launch(void* const* d_in, const int* in_sizes, int n_in,
                              void* d_out, int out_size, void* d_ws, size_t ws_size,
                              hipStream_t stream) {
  const float* x = (const float*)d_in[0];
  const float* w = (const float*)d_in[1];
  float* y = (float*)d_out;
  dim3 grid(kP / kPB, kN / 16);             // (16, 512) workgroups of 128 threads (4 waves)
  LocBlock1dNT_wmma_kernel<<<grid, dim3(128), 0, stream>>>(x, w, y);
}